// BEVPool_7069516169822
// MI455X (gfx1250) — compile-verified
//
#include <hip/hip_runtime.h>

// BEVPool scatter-add for MI455X (gfx1250, wave32).
// Pass 1: zero accumulator. Pass 2: voxelize + global_atomic_add_f32 scatter
// (channel-minor layout when workspace available). Pass 3: LDS transpose to
// channel-major output with non-temporal stores.

typedef float f32x4 __attribute__((ext_vector_type(4)));

#define HGRID 360
#define WGRID 360
#define HWGRID (HGRID * WGRID)   // 129600 voxels (D=1, B=1)
#define CCH 64

// ---------------------------------------------------------------- zero fill
__global__ void bev_zero_f4(f32x4* __restrict__ p, int n4) {
    int i = blockIdx.x * blockDim.x + threadIdx.x;
    if (i < n4) {
        f32x4 z = {0.f, 0.f, 0.f, 0.f};
        p[i] = z;
    }
}

// ------------------------------------------------- native f32 atomic add
// Non-returning global_atomic_add_f32 (tracked on STOREcnt; s_endpgm's
// implicit wait-idle guarantees completion before kernel end).
__device__ __forceinline__ void gatomic_add_f32(float* p, float v) {
    asm volatile("global_atomic_add_f32 %0, %1, off"
                 :
                 : "v"(p), "v"(v)
                 : "memory");
}

// ---------------------------------------------------------------- scatter
// 16 threads per point; each thread handles 4 consecutive channels via a
// non-temporal b128 load of x (one-touch stream: keep it out of L2 so the
// 33MB accumulator stays resident).
__global__ void bev_scatter(const float* __restrict__ geom,
                            const float* __restrict__ x,
                            float* __restrict__ accum,
                            int npoints, int stride_c, int stride_s) {
    int gid = blockIdx.x * blockDim.x + threadIdx.x;
    int p   = gid >> 4;
    if (p >= npoints) return;
    int c4 = (gid & 15) << 2;

    // voxelize: g = floor((geom - LO) / DX), LO=(-54,-54,-10), DX=(.3,.3,20)
    const float* g3 = geom + (size_t)p * 3;
    float fx = g3[0], fy = g3[1], fz = g3[2];
    int gx = (int)floorf((fx - (-54.0f)) / 0.3f);
    int gy = (int)floorf((fy - (-54.0f)) / 0.3f);
    int gz = (int)floorf((fz - (-10.0f)) / 20.0f);

    if (gx < 0 || gx >= HGRID || gy < 0 || gy >= WGRID || gz != 0) return;

    int spatial = gx * WGRID + gy;

    f32x4 f = __builtin_nontemporal_load(
        (const f32x4*)(x + (size_t)p * CCH + c4));

    float* base = accum + (size_t)spatial * (size_t)stride_s
                        + (size_t)c4      * (size_t)stride_c;
    gatomic_add_f32(base + 0 * stride_c, f.x);
    gatomic_add_f32(base + 1 * stride_c, f.y);
    gatomic_add_f32(base + 2 * stride_c, f.z);
    gatomic_add_f32(base + 3 * stride_c, f.w);
}

// -------------------------------------------------------------- transpose
// in: [HW][64] channel-minor accumulator -> out: [64][HW] channel-major.
// 64x64 f32 tile in LDS with +1 padding (bank-conflict free), vectorized
// b128 on both sides, NT stores for the final 33MB write.
__global__ void bev_transpose(const float* __restrict__ in,
                              float* __restrict__ out) {
    __shared__ float tile[64][65];
    int row0 = blockIdx.x * 64;
    int t    = threadIdx.x;            // 256 threads = 8 wave32

    // load 64 rows x 64 ch, coalesced b128
    #pragma unroll
    for (int k = 0; k < 4; ++k) {
        int l  = t + k * 256;          // 0..1023 float4 slots
        int r  = l >> 4;               // 0..63
        int c4 = (l & 15) << 2;        // 0..60
        f32x4 v = *(const f32x4*)(in + (size_t)(row0 + r) * CCH + c4);
        tile[r][c4 + 0] = v.x;
        tile[r][c4 + 1] = v.y;
        tile[r][c4 + 2] = v.z;
        tile[r][c4 + 3] = v.w;
    }
    __syncthreads();

    // store transposed: contiguous along spatial dim per channel
    #pragma unroll
    for (int k = 0; k < 4; ++k) {
        int l  = t + k * 256;
        int c  = l >> 4;               // channel 0..63
        int r4 = (l & 15) << 2;        // row group 0..60
        f32x4 v;
        v.x = tile[r4 + 0][c];
        v.y = tile[r4 + 1][c];
        v.z = tile[r4 + 2][c];
        v.w = tile[r4 + 3][c];
        __builtin_nontemporal_store(
            v, (f32x4*)(out + (size_t)c * HWGRID + row0 + r4));
    }
}

// ------------------------------------------------------------------ launch
extern "C" void kernel_launch(void* const* d_in, const int* in_sizes, int n_in,
                              void* d_out, int out_size, void* d_ws, size_t ws_size,
                              hipStream_t stream) {
    const float* geom = (const float*)d_in[0];   // [Np,3]
    const float* x    = (const float*)d_in[1];   // [Np,64]
    float* out        = (float*)d_out;           // [64,360,360]

    const int npoints = in_sizes[1] / CCH;       // 1,993,728
    const size_t accum_bytes = (size_t)HWGRID * CCH * sizeof(float);

    const int scatter_threads = npoints * 16;
    const int scatter_blocks  = (scatter_threads + 255) / 256;

    if (ws_size >= accum_bytes) {
        // two-pass: channel-minor accumulate in ws, then transpose to out
        float* accum = (float*)d_ws;
        int n4 = (HWGRID * CCH) / 4;             // 2,073,600
        bev_zero_f4<<<(n4 + 255) / 256, 256, 0, stream>>>((f32x4*)accum, n4);
        bev_scatter<<<scatter_blocks, 256, 0, stream>>>(
            geom, x, accum, npoints, /*stride_c=*/1, /*stride_s=*/CCH);
        bev_transpose<<<HWGRID / 64, 256, 0, stream>>>(accum, out);
    } else {
        // fallback: direct strided atomics into channel-major output
        int n4 = (HWGRID * CCH) / 4;
        bev_zero_f4<<<(n4 + 255) / 256, 256, 0, stream>>>((f32x4*)out, n4);
        bev_scatter<<<scatter_blocks, 256, 0, stream>>>(
            geom, x, out, npoints, /*stride_c=*/HWGRID, /*stride_s=*/1);
    }
}